// GATNet_37177236914409
// MI455X (gfx1250) — compile-verified
//
#include <hip/hip_runtime.h>
#include <hip/hip_bf16.h>
#include <math.h>

#define NEG_SLOPE 0.2f

typedef __attribute__((ext_vector_type(2))) float v2f;
typedef __attribute__((ext_vector_type(8))) float v8f;

// ---------------------------------------------------------------------------
// Monotone float <-> uint mapping so atomicMax(u32) implements float max.
// ---------------------------------------------------------------------------
__device__ __forceinline__ unsigned fmap(float f) {
  unsigned u = __float_as_uint(f);
  return (u & 0x80000000u) ? ~u : (u | 0x80000000u);
}
__device__ __forceinline__ float funmap(unsigned u) {
  unsigned v = (u & 0x80000000u) ? (u ^ 0x80000000u) : ~u;
  return __uint_as_float(v);
}
// mapped(-inf) = ~0xFF800000 = 0x007FFFFF
#define FMAP_NEG_INF 0x007FFFFFu

// ---------------------------------------------------------------------------
// Dense GEMM  H[M,NC] = X[M,K] @ W[K,NC]  via V_WMMA_F32_16X16X4_F32.
// One wave computes one 16x16 tile: blockIdx.x = M-tile, wave-id = N-tile.
// Requires M % 16 == 0 (true: 50000 = 3125*16) and NC = 16 * warps/block.
// A frag (16x4): lane m = lane&15, VGPR0 -> K=kb, VGPR1 -> K=kb+1, kb=2*(lane>>4)
// B frag (4x16): lane n = lane&15, same K striping.
// D frag: VGPR v holds row (v + 8*(lane>>4)), col = lane&15.
// ---------------------------------------------------------------------------
template <int K, int NC>
__global__ __launch_bounds__(256) void gemm_wmma_f32(
    const float* __restrict__ X, const float* __restrict__ W,
    float* __restrict__ H) {
  const int lane  = threadIdx.x & 31;
  const int wave  = threadIdx.x >> 5;      // N-tile index
  const int mtile = blockIdx.x;
  const int mloc  = lane & 15;
  const int kb    = (lane >> 4) << 1;      // 0 or 2
  const int rowA  = mtile * 16 + mloc;
  const int ncol  = wave * 16 + mloc;

  v8f acc0 = {};
  v8f acc1 = {};
#pragma unroll
  for (int k0 = 0; k0 < K; k0 += 8) {
    // ---- step 0: K slice [k0, k0+4) ----
    {
      v2f a, b;
      const float* ap = X + (size_t)rowA * K + k0 + kb;
      a.x = ap[0];
      a.y = ap[1];
      b.x = W[(size_t)(k0 + kb) * NC + ncol];
      b.y = W[(size_t)(k0 + kb + 1) * NC + ncol];
      acc0 = __builtin_amdgcn_wmma_f32_16x16x4_f32(false, a, false, b,
                                                   (short)0, acc0, false, false);
    }
    // ---- step 1: K slice [k0+4, k0+8) ----
    {
      v2f a, b;
      const float* ap = X + (size_t)rowA * K + k0 + 4 + kb;
      a.x = ap[0];
      a.y = ap[1];
      b.x = W[(size_t)(k0 + 4 + kb) * NC + ncol];
      b.y = W[(size_t)(k0 + 5 + kb) * NC + ncol];
      acc1 = __builtin_amdgcn_wmma_f32_16x16x4_f32(false, a, false, b,
                                                   (short)0, acc1, false, false);
    }
  }

  const int rhalf = (lane >> 4) * 8;
#pragma unroll
  for (int v = 0; v < 8; ++v) {
    int row = mtile * 16 + v + rhalf;
    H[(size_t)row * NC + ncol] = acc0[v] + acc1[v];
  }
}

// ---------------------------------------------------------------------------
// Per-(node,head) attention coefficients, layer 1 (8 heads x 16 ch).
// ---------------------------------------------------------------------------
__global__ __launch_bounds__(256) void alpha_heads8(
    const float* __restrict__ h, const float* __restrict__ a_src,
    const float* __restrict__ a_dst, float* __restrict__ as,
    float* __restrict__ ad, int N) {
  int i = blockIdx.x * blockDim.x + threadIdx.x;
  if (i >= N * 8) return;
  int n = i >> 3, hd = i & 7;
  const float4* hv = (const float4*)(h + (size_t)n * 128 + hd * 16);
  const float4* sv = (const float4*)(a_src + hd * 16);
  const float4* dv = (const float4*)(a_dst + hd * 16);
  float sa = 0.f, da = 0.f;
#pragma unroll
  for (int c = 0; c < 4; ++c) {
    float4 v = hv[c], s = sv[c], d = dv[c];
    sa += v.x * s.x + v.y * s.y + v.z * s.z + v.w * s.w;
    da += v.x * d.x + v.y * d.y + v.z * d.z + v.w * d.w;
  }
  as[i] = sa;
  ad[i] = da;
}

// Layer 2 (1 head x 64 ch), one thread per node.
__global__ __launch_bounds__(256) void alpha_head1(
    const float* __restrict__ h, const float* __restrict__ a_src,
    const float* __restrict__ a_dst, float* __restrict__ as,
    float* __restrict__ ad, int N) {
  int n = blockIdx.x * blockDim.x + threadIdx.x;
  if (n >= N) return;
  const float4* hv = (const float4*)(h + (size_t)n * 64);
  const float4* sv = (const float4*)a_src;
  const float4* dv = (const float4*)a_dst;
  float sa = 0.f, da = 0.f;
#pragma unroll
  for (int c = 0; c < 16; ++c) {
    float4 v = hv[c], s = sv[c], d = dv[c];
    sa += v.x * s.x + v.y * s.y + v.z * s.z + v.w * s.w;
    da += v.x * d.x + v.y * d.y + v.z * d.z + v.w * d.w;
  }
  as[n] = sa;
  ad[n] = da;
}

__global__ __launch_bounds__(256) void fill_u32(unsigned* __restrict__ p,
                                                unsigned val, int n) {
  int i = blockIdx.x * blockDim.x + threadIdx.x;
  if (i < n) p[i] = val;
}

// ---------------------------------------------------------------------------
// Edge passes, layer 1 (8 heads). Edges [0,E) from index arrays, [E,Etot)
// are self-loops (src = dst = e - E), matching the reference concat order.
// ---------------------------------------------------------------------------
__global__ __launch_bounds__(256) void edge_max8(
    const int* __restrict__ src, const int* __restrict__ dst, int E, int Etot,
    const float* __restrict__ as, const float* __restrict__ ad,
    unsigned* __restrict__ m) {
  int e = blockIdx.x * blockDim.x + threadIdx.x;
  if (e >= Etot) return;
  int s = (e < E) ? src[e] : (e - E);
  int d = (e < E) ? dst[e] : (e - E);
  const float4* av = (const float4*)(as + (size_t)s * 8);
  const float4* bv = (const float4*)(ad + (size_t)d * 8);
  float4 a0 = av[0], a1 = av[1], b0 = bv[0], b1 = bv[1];
  float ev[8] = {a0.x + b0.x, a0.y + b0.y, a0.z + b0.z, a0.w + b0.w,
                 a1.x + b1.x, a1.y + b1.y, a1.z + b1.z, a1.w + b1.w};
#pragma unroll
  for (int hd = 0; hd < 8; ++hd) {
    float t = ev[hd];
    t = t > 0.f ? t : NEG_SLOPE * t;
    atomicMax(&m[(size_t)d * 8 + hd], fmap(t));
  }
}

__global__ __launch_bounds__(256) void edge_sum8(
    const int* __restrict__ src, const int* __restrict__ dst, int E, int Etot,
    const float* __restrict__ as, const float* __restrict__ ad,
    const unsigned* __restrict__ m, float* __restrict__ ssum) {
  int e = blockIdx.x * blockDim.x + threadIdx.x;
  if (e >= Etot) return;
  int s = (e < E) ? src[e] : (e - E);
  int d = (e < E) ? dst[e] : (e - E);
  const float4* av = (const float4*)(as + (size_t)s * 8);
  const float4* bv = (const float4*)(ad + (size_t)d * 8);
  float4 a0 = av[0], a1 = av[1], b0 = bv[0], b1 = bv[1];
  float ev[8] = {a0.x + b0.x, a0.y + b0.y, a0.z + b0.z, a0.w + b0.w,
                 a1.x + b1.x, a1.y + b1.y, a1.z + b1.z, a1.w + b1.w};
#pragma unroll
  for (int hd = 0; hd < 8; ++hd) {
    float t = ev[hd];
    t = t > 0.f ? t : NEG_SLOPE * t;
    float mm = funmap(m[(size_t)d * 8 + hd]);
    if (mm < -3.0e38f) mm = 0.f;  // isfinite guard (self-loops make m finite)
    atomicAdd(&ssum[(size_t)d * 8 + hd], __expf(t - mm));
  }
}

// One thread per (edge, head): 16-channel weighted scatter-add.
__global__ __launch_bounds__(256) void edge_aggr8(
    const int* __restrict__ src, const int* __restrict__ dst, int E, int Etot,
    const float* __restrict__ as, const float* __restrict__ ad,
    const unsigned* __restrict__ m, const float* __restrict__ ssum,
    const float* __restrict__ h, float* __restrict__ out) {
  int i = blockIdx.x * blockDim.x + threadIdx.x;
  int e = i >> 3, hd = i & 7;
  if (e >= Etot) return;
  int s = (e < E) ? src[e] : (e - E);
  int d = (e < E) ? dst[e] : (e - E);
  float t = as[(size_t)s * 8 + hd] + ad[(size_t)d * 8 + hd];
  t = t > 0.f ? t : NEG_SLOPE * t;
  float mm = funmap(m[(size_t)d * 8 + hd]);
  if (mm < -3.0e38f) mm = 0.f;
  float alpha = __expf(t - mm) / (ssum[(size_t)d * 8 + hd] + 1e-16f);
  const float4* hv = (const float4*)(h + (size_t)s * 128 + hd * 16);
  float* op = out + (size_t)d * 128 + hd * 16;
#pragma unroll
  for (int c = 0; c < 4; ++c) {
    float4 v = hv[c];
    atomicAdd(op + c * 4 + 0, alpha * v.x);
    atomicAdd(op + c * 4 + 1, alpha * v.y);
    atomicAdd(op + c * 4 + 2, alpha * v.z);
    atomicAdd(op + c * 4 + 3, alpha * v.w);
  }
}

// out1 = elu(out1 + b1)
__global__ __launch_bounds__(256) void bias_elu128(float* __restrict__ out,
                                                   const float* __restrict__ b,
                                                   int total) {
  int i = blockIdx.x * blockDim.x + threadIdx.x;
  if (i >= total) return;
  float v = out[i] + b[i & 127];
  out[i] = v > 0.f ? v : (__expf(v) - 1.f);
}

// ---------------------------------------------------------------------------
// Edge passes, layer 2 (1 head x 64 ch).
// ---------------------------------------------------------------------------
__global__ __launch_bounds__(256) void edge_max1(
    const int* __restrict__ src, const int* __restrict__ dst, int E, int Etot,
    const float* __restrict__ as, const float* __restrict__ ad,
    unsigned* __restrict__ m) {
  int e = blockIdx.x * blockDim.x + threadIdx.x;
  if (e >= Etot) return;
  int s = (e < E) ? src[e] : (e - E);
  int d = (e < E) ? dst[e] : (e - E);
  float t = as[s] + ad[d];
  t = t > 0.f ? t : NEG_SLOPE * t;
  atomicMax(&m[d], fmap(t));
}

__global__ __launch_bounds__(256) void edge_sum1(
    const int* __restrict__ src, const int* __restrict__ dst, int E, int Etot,
    const float* __restrict__ as, const float* __restrict__ ad,
    const unsigned* __restrict__ m, float* __restrict__ ssum) {
  int e = blockIdx.x * blockDim.x + threadIdx.x;
  if (e >= Etot) return;
  int s = (e < E) ? src[e] : (e - E);
  int d = (e < E) ? dst[e] : (e - E);
  float t = as[s] + ad[d];
  t = t > 0.f ? t : NEG_SLOPE * t;
  float mm = funmap(m[d]);
  if (mm < -3.0e38f) mm = 0.f;
  atomicAdd(&ssum[d], __expf(t - mm));
}

// One thread per (edge, half): 32-channel weighted scatter-add.
__global__ __launch_bounds__(256) void edge_aggr1(
    const int* __restrict__ src, const int* __restrict__ dst, int E, int Etot,
    const float* __restrict__ as, const float* __restrict__ ad,
    const unsigned* __restrict__ m, const float* __restrict__ ssum,
    const float* __restrict__ h, float* __restrict__ out) {
  int i = blockIdx.x * blockDim.x + threadIdx.x;
  int e = i >> 1, half = i & 1;
  if (e >= Etot) return;
  int s = (e < E) ? src[e] : (e - E);
  int d = (e < E) ? dst[e] : (e - E);
  float t = as[s] + ad[d];
  t = t > 0.f ? t : NEG_SLOPE * t;
  float mm = funmap(m[d]);
  if (mm < -3.0e38f) mm = 0.f;
  float alpha = __expf(t - mm) / (ssum[d] + 1e-16f);
  const float4* hv = (const float4*)(h + (size_t)s * 64 + half * 32);
  float* op = out + (size_t)d * 64 + half * 32;
#pragma unroll
  for (int c = 0; c < 8; ++c) {
    float4 v = hv[c];
    atomicAdd(op + c * 4 + 0, alpha * v.x);
    atomicAdd(op + c * 4 + 1, alpha * v.y);
    atomicAdd(op + c * 4 + 2, alpha * v.z);
    atomicAdd(op + c * 4 + 3, alpha * v.w);
  }
}

__global__ __launch_bounds__(256) void bias_out64(float* __restrict__ out,
                                                  const float* __restrict__ b,
                                                  int total) {
  int i = blockIdx.x * blockDim.x + threadIdx.x;
  if (i >= total) return;
  out[i] += b[i & 63];
}

// ---------------------------------------------------------------------------
// Host-side orchestration (graph-capture safe: only kernels + memsetAsync).
// ---------------------------------------------------------------------------
extern "C" void kernel_launch(void* const* d_in, const int* in_sizes, int n_in,
                              void* d_out, int out_size, void* d_ws,
                              size_t ws_size, hipStream_t stream) {
  const float* x   = (const float*)d_in[0];
  const int*   ei  = (const int*)d_in[1];
  const float* W1  = (const float*)d_in[2];
  const float* a1s = (const float*)d_in[3];
  const float* a1d = (const float*)d_in[4];
  const float* b1  = (const float*)d_in[5];
  const float* W2  = (const float*)d_in[6];
  const float* a2s = (const float*)d_in[7];
  const float* a2d = (const float*)d_in[8];
  const float* b2  = (const float*)d_in[9];
  float* out = (float*)d_out;

  const int N    = in_sizes[0] / 128;  // 50000 (multiple of 16)
  const int E    = in_sizes[1] / 2;    // 800000
  const int Etot = E + N;              // edges + self-loops
  const int* src = ei;
  const int* dst = ei + E;

  // -------- workspace carve-up (256B aligned) --------
  char* base = (char*)d_ws;
  size_t off = 0;
  auto alloc = [&](size_t bytes) -> void* {
    void* p = base + off;
    off += (bytes + 255) & ~(size_t)255;
    return p;
  };
  float*    h1   = (float*)alloc((size_t)N * 128 * 4);
  float*    out1 = (float*)alloc((size_t)N * 128 * 4);
  float*    h2   = (float*)alloc((size_t)N * 64 * 4);
  float*    as1  = (float*)alloc((size_t)N * 8 * 4);
  float*    ad1  = (float*)alloc((size_t)N * 8 * 4);
  unsigned* m1   = (unsigned*)alloc((size_t)N * 8 * 4);
  float*    s1   = (float*)alloc((size_t)N * 8 * 4);
  float*    as2  = (float*)alloc((size_t)N * 4);
  float*    ad2  = (float*)alloc((size_t)N * 4);
  unsigned* m2   = (unsigned*)alloc((size_t)N * 4);
  float*    s2   = (float*)alloc((size_t)N * 4);

  const int TB = 256;
  auto blocks = [&](long long n) { return (int)((n + TB - 1) / TB); };

  // ================= Layer 1 =================
  // h1 = x @ W1   (WMMA f32, 8 head-tiles per block = 8 waves)
  gemm_wmma_f32<128, 128><<<N / 16, 256, 0, stream>>>(x, W1, h1);
  // attention coefficients
  alpha_heads8<<<blocks((long long)N * 8), TB, 0, stream>>>(h1, a1s, a1d, as1,
                                                            ad1, N);
  // init segment stats + output accumulator
  fill_u32<<<blocks((long long)N * 8), TB, 0, stream>>>(m1, FMAP_NEG_INF,
                                                        N * 8);
  hipMemsetAsync(s1, 0, (size_t)N * 8 * 4, stream);
  hipMemsetAsync(out1, 0, (size_t)N * 128 * 4, stream);
  // segment softmax + aggregate
  edge_max8<<<blocks(Etot), TB, 0, stream>>>(src, dst, E, Etot, as1, ad1, m1);
  edge_sum8<<<blocks(Etot), TB, 0, stream>>>(src, dst, E, Etot, as1, ad1, m1,
                                             s1);
  edge_aggr8<<<blocks((long long)Etot * 8), TB, 0, stream>>>(
      src, dst, E, Etot, as1, ad1, m1, s1, h1, out1);
  // + bias, ELU
  bias_elu128<<<blocks((long long)N * 128), TB, 0, stream>>>(out1, b1,
                                                             N * 128);

  // ================= Layer 2 =================
  // h2 = out1 @ W2   (WMMA f32, 4 tiles per block = 4 waves)
  gemm_wmma_f32<128, 64><<<N / 16, 128, 0, stream>>>(out1, W2, h2);
  alpha_head1<<<blocks(N), TB, 0, stream>>>(h2, a2s, a2d, as2, ad2, N);
  fill_u32<<<blocks(N), TB, 0, stream>>>(m2, FMAP_NEG_INF, N);
  hipMemsetAsync(s2, 0, (size_t)N * 4, stream);
  hipMemsetAsync(out, 0, (size_t)N * 64 * 4, stream);
  edge_max1<<<blocks(Etot), TB, 0, stream>>>(src, dst, E, Etot, as2, ad2, m2);
  edge_sum1<<<blocks(Etot), TB, 0, stream>>>(src, dst, E, Etot, as2, ad2, m2,
                                             s2);
  edge_aggr1<<<blocks((long long)Etot * 2), TB, 0, stream>>>(
      src, dst, E, Etot, as2, ad2, m2, s2, h2, out);
  bias_out64<<<blocks((long long)N * 64), TB, 0, stream>>>(out, b2, N * 64);
}